// PaddedProbAttention_52037823758871
// MI455X (gfx1250) — compile-verified
//
#include <hip/hip_runtime.h>

// ProbSparse attention (Informer) for MI455X / gfx1250.
// B=4 H=8 L_Q=L_K=2048 D=64, sample_k = n_top = 40.
// Output = dense [B,H,LQ,LK] fp32 softmax matrix (512 MiB -> store-bandwidth bound).

#define B_    4
#define H_    8
#define BH_   (B_ * H_)      // 32
#define LQ_   2048
#define LK_   2048
#define D_    64
#define NTOP  40
#define SK    40
#define SCALE 0.125f         // 1/sqrt(64)

typedef __attribute__((ext_vector_type(2))) float v2f;
typedef __attribute__((ext_vector_type(4))) float v4f;
typedef __attribute__((ext_vector_type(8))) float v8f;

// ---------------------------------------------------------------------------
// Kernel 1: M[b,h,l] = max_s(Q.K_sample) - sum_s(Q.K_sample)/L_K
// One wave per query row; lane s handles sampled keys {s, s+32} (s<40).
// ---------------------------------------------------------------------------
__global__ void pk_sample_m(const float* __restrict__ Q,
                            const float* __restrict__ K,
                            const int*   __restrict__ idxs,
                            float*       __restrict__ Mout) {
  __shared__ float qsh[8][D_];
  const int w    = threadIdx.x >> 5;
  const int lane = threadIdx.x & 31;
  const int r    = blockIdx.x * 8 + w;       // row in [0, B*H*LQ)
  const int bh   = r / LQ_;
  const int l    = r % LQ_;
  const int b    = bh / H_, h = bh % H_;

  const float* qrow = Q + (((size_t)b * LQ_ + l) * H_ + h) * D_;
  qsh[w][lane]      = qrow[lane];
  qsh[w][lane + 32] = qrow[lane + 32];
  __syncthreads();

  float mx = -__builtin_inff();
  float sm = 0.0f;
  const v4f* q4 = (const v4f*)&qsh[w][0];
  for (int t = 0; t < 2; ++t) {
    const int s = lane + 32 * t;
    if (s < SK) {
      const int ki = idxs[l * SK + s];
      const v4f* k4 = (const v4f*)(K + (((size_t)b * LK_ + ki) * H_ + h) * D_);
      float acc = 0.0f;
      #pragma unroll
      for (int d4 = 0; d4 < D_ / 4; ++d4) {
        const v4f qq = q4[d4];
        const v4f kk = k4[d4];
        acc += qq.x * kk.x + qq.y * kk.y + qq.z * kk.z + qq.w * kk.w;
      }
      mx = fmaxf(mx, acc);
      sm += acc;
    }
  }
  #pragma unroll
  for (int off = 16; off >= 1; off >>= 1) {
    mx = fmaxf(mx, __shfl_xor(mx, off, 32));
    sm += __shfl_xor(sm, off, 32);
  }
  if (lane == 0) Mout[(size_t)bh * LQ_ + l] = mx - sm * (1.0f / (float)LK_);
}

// ---------------------------------------------------------------------------
// Kernel 2: per (b,h) top-40 of M (iterative argmax, lowest index on ties,
// matching jax.lax.top_k) + inverse row->slot map (0 = not selected).
// ---------------------------------------------------------------------------
__global__ void pk_topk(const float* __restrict__ Mvals,
                        int*         __restrict__ top,
                        int*         __restrict__ slot) {
  __shared__ float v[LQ_];
  __shared__ float rv[256];
  __shared__ int   ri[256];
  const int bh  = blockIdx.x;
  const int tid = threadIdx.x;

  for (int j = tid; j < LQ_; j += 256) {
    v[j] = Mvals[(size_t)bh * LQ_ + j];
    slot[(size_t)bh * LQ_ + j] = 0;         // re-zero every call (deterministic)
  }
  __syncthreads();

  for (int u = 0; u < NTOP; ++u) {
    float bv = -__builtin_inff();
    int   bi = 0x7fffffff;
    for (int j = tid; j < LQ_; j += 256) {
      const float x = v[j];
      if (x > bv || (x == bv && j < bi)) { bv = x; bi = j; }
    }
    rv[tid] = bv; ri[tid] = bi;
    __syncthreads();
    for (int s = 128; s > 0; s >>= 1) {
      if (tid < s) {
        const float ov = rv[tid + s]; const int oi = ri[tid + s];
        if (ov > rv[tid] || (ov == rv[tid] && oi < ri[tid])) { rv[tid] = ov; ri[tid] = oi; }
      }
      __syncthreads();
    }
    if (tid == 0) {
      const int win = ri[0];
      top[bh * NTOP + u] = win;
      slot[(size_t)bh * LQ_ + win] = u + 1;
      v[win] = -__builtin_inff();
    }
    __syncthreads();
  }
}

// ---------------------------------------------------------------------------
// Kernel 3: scores_top = (Q[M_top] @ K^T) * scale via V_WMMA_F32_16X16X4_F32,
// written straight into the selected output rows.
// Grid: BH*16 blocks x 256 thr (8 waves). Wave w handles col-tile cg*8+w,
// looping the 3 row-tiles (40 rows padded to 48).
// A frag (16x4 f32): lane m=lane&15 holds A[m][kb], A[m][kb+1], kb=4*kk+2*(lane>>4).
// B frag mirrors it: lane n=lane&15 holds B[kb][n], B[kb+1][n] = K[n][kb..kb+1].
// ---------------------------------------------------------------------------
__global__ void pk_scores_wmma(const float* __restrict__ Q,
                               const float* __restrict__ K,
                               const int*   __restrict__ top,
                               float*       __restrict__ out) {
  __shared__ float Qr[48][D_ + 4];   // +4 pad: A-frag reads spread over banks
  __shared__ int   itop[48];
  const int bh = blockIdx.x >> 4;
  const int cg = blockIdx.x & 15;
  const int b  = bh / H_, h = bh % H_;
  const int tid = threadIdx.x;

  if (tid < 48) itop[tid] = (tid < NTOP) ? top[bh * NTOP + tid] : 0;
  __syncthreads();

  for (int e = tid; e < 48 * D_; e += 256) {
    const int u = e >> 6, d = e & 63;
    float val = 0.0f;
    if (u < NTOP) val = Q[(((size_t)b * LQ_ + itop[u]) * H_ + h) * D_ + d];
    Qr[u][d] = val;
  }
  __syncthreads();

  const int w    = tid >> 5;
  const int lane = tid & 31;
  const int g    = lane >> 4;        // lane half-group
  const int n    = lane & 15;        // tile row (A) / tile col (B,D)
  const int ct   = cg * 8 + w;       // col tile 0..127
  const int kidx = ct * 16 + n;      // key index for this lane's B column
  const float* krow = K + (((size_t)b * LK_ + kidx) * H_ + h) * D_;

  for (int rt = 0; rt < 3; ++rt) {
    v8f c = {};
    const float* arow = &Qr[rt * 16 + n][0];
    #pragma unroll
    for (int kk = 0; kk < 16; ++kk) {
      const int kb = kk * 4 + g * 2;
      v2f a;  a.x  = arow[kb]; a.y  = arow[kb + 1];
      v2f bb; bb.x = krow[kb]; bb.y = krow[kb + 1];
      c = __builtin_amdgcn_wmma_f32_16x16x4_f32(
          /*neg_a=*/false, a, /*neg_b=*/false, bb,
          /*c_mod=*/(short)0, c, /*reuse_a=*/false, /*reuse_b=*/false);
    }
    #pragma unroll
    for (int rl = 0; rl < 8; ++rl) {
      const int u = rt * 16 + rl + 8 * g;   // D layout: vgpr rl -> row rl + 8*g
      if (u < NTOP) {
        const int i = itop[u];
        out[((size_t)bh * LQ_ + i) * LK_ + ct * 16 + n] = c[rl] * SCALE;
      }
    }
  }
}

// ---------------------------------------------------------------------------
// Kernel 4: in-place causal softmax of the 1280 selected rows.
// ---------------------------------------------------------------------------
__global__ void pk_softmax_rows(const int* __restrict__ top,
                                float*     __restrict__ out) {
  __shared__ float red[256];
  const int bh = blockIdx.x / NTOP;
  const int u  = blockIdx.x % NTOP;
  const int i  = top[bh * NTOP + u];
  float* row = out + ((size_t)bh * LQ_ + i) * LK_;
  const int tid = threadIdx.x;

  float lm = -__builtin_inff();
  for (int j = tid; j < LK_; j += 256)
    if (j <= i) lm = fmaxf(lm, row[j]);
  red[tid] = lm; __syncthreads();
  for (int s = 128; s > 0; s >>= 1) {
    if (tid < s) red[tid] = fmaxf(red[tid], red[tid + s]);
    __syncthreads();
  }
  const float smax = red[0];
  __syncthreads();

  float ls = 0.0f;
  for (int j = tid; j < LK_; j += 256)
    if (j <= i) ls += __expf(row[j] - smax);
  red[tid] = ls; __syncthreads();
  for (int s = 128; s > 0; s >>= 1) {
    if (tid < s) red[tid] += red[tid + s];
    __syncthreads();
  }
  const float inv = 1.0f / red[0];

  for (int j = tid; j < LK_; j += 256)
    row[j] = (j <= i) ? __expf(row[j] - smax) * inv : 0.0f;
}

// ---------------------------------------------------------------------------
// Kernel 5: stream the ~64K non-selected rows: 1/(i+1) up to the diagonal.
// Non-temporal ext-vector float4 stores: 512 MiB write-once traffic,
// keep it out of cache (TH=NT).
// ---------------------------------------------------------------------------
__global__ void pk_fill_uniform(const int* __restrict__ slot,
                                float*     __restrict__ out) {
  const int row = blockIdx.x;                 // bh*LQ + i
  if (slot[row] != 0) return;                 // selected rows done by softmax
  const int i = row & (LQ_ - 1);
  const float val = 1.0f / (float)(i + 1);
  float* dst = out + (size_t)row * LK_;
  const int j0 = threadIdx.x * 8;
  #pragma unroll
  for (int q = 0; q < 2; ++q) {
    const int j = j0 + q * 4;
    v4f v;
    v.x = (j + 0 <= i) ? val : 0.0f;
    v.y = (j + 1 <= i) ? val : 0.0f;
    v.z = (j + 2 <= i) ? val : 0.0f;
    v.w = (j + 3 <= i) ? val : 0.0f;
    __builtin_nontemporal_store(v, (v4f*)(dst + j));
  }
}

// ---------------------------------------------------------------------------
extern "C" void kernel_launch(void* const* d_in, const int* in_sizes, int n_in,
                              void* d_out, int out_size, void* d_ws, size_t ws_size,
                              hipStream_t stream) {
  (void)in_sizes; (void)n_in; (void)out_size; (void)ws_size;
  const float* Q    = (const float*)d_in[0];   // [B,LQ,H,D]
  const float* K    = (const float*)d_in[1];   // [B,LK,H,D]
  // d_in[2] = values (unused by reference output), d_in[3] = attn_mask (unused)
  const int*   idxs = (const int*)d_in[4];     // [LQ, 40]
  float* out = (float*)d_out;                  // [B,H,LQ,LK]

  char* ws = (char*)d_ws;
  float* wsM    = (float*)ws;                                  // BH*LQ floats
  int*   wsTop  = (int*)(ws + (size_t)BH_ * LQ_ * 4);          // BH*NTOP ints
  int*   wsSlot = (int*)(ws + (size_t)BH_ * LQ_ * 4 + 5120);   // BH*LQ ints

  pk_sample_m    <<<BH_ * LQ_ / 8, 256, 0, stream>>>(Q, K, idxs, wsM);
  pk_topk        <<<BH_,           256, 0, stream>>>(wsM, wsTop, wsSlot);
  pk_scores_wmma <<<BH_ * 16,      256, 0, stream>>>(Q, K, wsTop, out);
  pk_softmax_rows<<<BH_ * NTOP,    256, 0, stream>>>(wsTop, out);
  pk_fill_uniform<<<BH_ * LQ_,     256, 0, stream>>>(wsSlot, out);
}